// RefineDetBase_73469710565613
// MI455X (gfx1250) — compile-verified
//
#include <hip/hip_runtime.h>
#include <hip/hip_bf16.h>
#include <stdint.h>

// ---------------- problem constants (match reference) ----------------
#define NB      8          // batch
#define NA      131072     // anchors (2^17)
#define NC      21         // classes incl. background
#define NCLS    20         // foreground classes
#define NPAIR   (NB*NCLS)  // 160 independent (b,c) problems
#define NCAND   1000       // top-k candidates
#define CANDPAD 1024       // padded stride for candidate buffers
#define MAXOUT  200        // NMS outputs per (b,c)
#define CONF_T  0.01f
#define ARM_T   0.99f
#define NMS_T   0.45f
#define VAR_XY  0.1f
#define VAR_WH  0.2f

#define HAS_ASYNC_LDS (__has_builtin(__builtin_amdgcn_global_load_async_to_lds_b32) && \
                       __has_builtin(__builtin_amdgcn_global_load_async_to_lds_b128))

// exact parameter types per hipcc diagnostics:
//   b32 : (addrspace(1) int*,  addrspace(3) int*,  imm int, imm int)
//   b128: (addrspace(1) v4i*,  addrspace(3) v4i*,  imm int, imm int)
typedef __attribute__((ext_vector_type(4))) int v4i;
typedef __attribute__((address_space(1))) int  g_int;
typedef __attribute__((address_space(3))) int  s_int;
typedef __attribute__((address_space(1))) v4i  g_v4i;
typedef __attribute__((address_space(3))) v4i  s_v4i;

// ---------------------------------------------------------------------
// Kernel 1: fused ARM+ODM decode, ignore-mask, threshold, transpose.
// 1 thread per (b,a). Writes boxes (B,A,4) and scoresT (B*20, A).
// ---------------------------------------------------------------------
__global__ __launch_bounds__(256) void decode_kernel(
    const float* __restrict__ arm_cls, const float* __restrict__ arm_loc,
    const float* __restrict__ odm_cls, const float* __restrict__ odm_loc,
    const float* __restrict__ anchors, float* __restrict__ boxes,
    float* __restrict__ scoresT)
{
    int i = blockIdx.x * blockDim.x + threadIdx.x;   // 0 .. NB*NA-1
    if (i >= NB * NA) return;
    int b = i >> 17;            // NA == 2^17
    int a = i & (NA - 1);

    const float* oc = odm_cls + (size_t)i * NC;
    __builtin_prefetch(oc, 0, 1);                    // global_prefetch_b8

    float4 an = ((const float4*)anchors)[a];
    float4 al = ((const float4*)arm_loc)[i];
    float4 ol = ((const float4*)odm_loc)[i];

    // ARM decode: refined box center/size (algebraically equals decode->re-center)
    float aw  = an.z - an.x;
    float ah  = an.w - an.y;
    float acx = an.x + 0.5f * aw;
    float acy = an.y + 0.5f * ah;
    float cx  = al.x * VAR_XY * aw + acx;
    float cy  = al.y * VAR_XY * ah + acy;
    float w   = aw * expf(al.z * VAR_WH);
    float h   = ah * expf(al.w * VAR_WH);

    // ODM decode on refined anchor
    float bcx = ol.x * VAR_XY * w + cx;
    float bcy = ol.y * VAR_XY * h + cy;
    float bw  = w * expf(ol.z * VAR_WH);
    float bh  = h * expf(ol.w * VAR_WH);

    float4 bx;
    bx.x = bcx - 0.5f * bw;
    bx.y = bcy - 0.5f * bh;
    bx.z = bcx + 0.5f * bw;
    bx.w = bcy + 0.5f * bh;
    ((float4*)boxes)[i] = bx;

    // ARM ignore + confidence threshold, transposed class-major store
    float keep = (arm_cls[2 * i] >= ARM_T) ? 0.0f : 1.0f;
#pragma unroll
    for (int c = 1; c < NC; ++c) {
        float s = oc[c] * keep;
        s = (s > CONF_T) ? s : 0.0f;
        scoresT[(((size_t)(b * NCLS + (c - 1))) << 17) + a] = s;
    }
}

// ---------------------------------------------------------------------
// Kernel 2: exact top-1000 per (b,c) via 2-level 11-bit radix select on
// float bit patterns (all scores >= 0 -> bits are order-preserving),
// then compaction of candidates (score + box) into a 1000-slot buffer.
// ---------------------------------------------------------------------
__global__ __launch_bounds__(256) void select_kernel(
    const float* __restrict__ scoresT, const float* __restrict__ boxes,
    float* __restrict__ candS, float* __restrict__ candB)
{
    int p = blockIdx.x;            // (b*20 + c) pair
    int t = threadIdx.x;
    int b = p / NCLS;
    const float* sp = scoresT + ((size_t)p << 17);

    __shared__ unsigned hist[2048];
    __shared__ int sB1, sNeed1, sB2, sNeed2, sAboveTot;
    __shared__ int cAbove, cTie;

    // ---- pass 1: histogram of top 11 bits ----
    for (int i = t; i < 2048; i += 256) hist[i] = 0u;
    __syncthreads();
    for (int i = t; i < NA; i += 256) {
        unsigned u = __float_as_uint(sp[i]);
        atomicAdd(&hist[u >> 21], 1u);
    }
    __syncthreads();
    if (t == 0) {
        unsigned cum = 0; int bin = 0;
        for (int i = 2047; i >= 0; --i) {
            unsigned c = hist[i];
            if (cum + c >= (unsigned)NCAND) { bin = i; break; }
            cum += c;
        }
        sB1 = bin; sNeed1 = NCAND - (int)cum;
    }
    __syncthreads();
    int B1 = sB1, need1 = sNeed1;

    // ---- pass 2: histogram of next 11 bits within bin B1 ----
    for (int i = t; i < 2048; i += 256) hist[i] = 0u;
    __syncthreads();
    for (int i = t; i < NA; i += 256) {
        unsigned u = __float_as_uint(sp[i]);
        if ((int)(u >> 21) == B1) atomicAdd(&hist[(u >> 10) & 2047u], 1u);
    }
    __syncthreads();
    if (t == 0) {
        unsigned cum = 0; int bin = 0;
        for (int i = 2047; i >= 0; --i) {
            unsigned c = hist[i];
            if (cum + c >= (unsigned)need1) { bin = i; break; }
            cum += c;
        }
        sB2 = bin; sNeed2 = need1 - (int)cum;
        sAboveTot = NCAND - sNeed2;     // exact count of strictly-greater keys
        cAbove = 0; cTie = 0;
    }
    __syncthreads();

    unsigned key = ((unsigned)B1 << 11) | (unsigned)sB2;
    int need2 = sNeed2, aboveTot = sAboveTot;
    const float4* bxp = (const float4*)boxes + ((size_t)b << 17);
    float*  cs = candS + (size_t)p * CANDPAD;
    float4* cb = (float4*)candB + (size_t)p * CANDPAD;

    // ---- pass 3: compaction (exactly 1000 slots filled) ----
    for (int i = t; i < NA; i += 256) {
        float v = sp[i];
        unsigned k12 = __float_as_uint(v) >> 10;
        int slot = -1;
        if (k12 > key) {
            slot = atomicAdd(&cAbove, 1);
        } else if (k12 == key) {
            int tt = atomicAdd(&cTie, 1);
            if (tt < need2) slot = aboveTot + tt;
        }
        if (slot >= 0) { cs[slot] = v; cb[slot] = bxp[i]; }
    }
}

// ---------------------------------------------------------------------
// Kernel 3: greedy NMS per (b,c). Candidates staged into LDS with CDNA5
// async global->LDS loads; 200 rounds of wave32-shuffle argmax + IoU
// suppression, output rows [score, x1,y1,x2,y2], zero-padded on exit.
// ---------------------------------------------------------------------
__global__ __launch_bounds__(256) void nms_kernel(
    const float* __restrict__ candS, const float* __restrict__ candB,
    float* __restrict__ out)
{
    int p = blockIdx.x;
    int t = threadIdx.x;

    __shared__ float  s_sc[NCAND];
    __shared__ float  s_ar[NCAND];
    __shared__ float4 s_bx[NCAND];
    __shared__ float  rS[8];
    __shared__ int    rJ[8];
    __shared__ float  sBest;
    __shared__ int    sBestJ;

    const float*  cs = candS + (size_t)p * CANDPAD;
    const float4* cb = (const float4*)candB + (size_t)p * CANDPAD;

#if HAS_ASYNC_LDS
    for (int k = t; k < NCAND; k += 256) {
        __builtin_amdgcn_global_load_async_to_lds_b32(
            (g_int*)(cs + k), (s_int*)&s_sc[k], 0, 0);
        __builtin_amdgcn_global_load_async_to_lds_b128(
            (g_v4i*)(cb + k), (s_v4i*)&s_bx[k], 0, 0);
    }
#if __has_builtin(__builtin_amdgcn_s_wait_asynccnt)
    __builtin_amdgcn_s_wait_asynccnt(0);
#else
    asm volatile("s_wait_asynccnt 0" ::: "memory");
#endif
#else
    for (int k = t; k < NCAND; k += 256) {
        s_sc[k] = cs[k];
        s_bx[k] = cb[k];
    }
#endif
    __syncthreads();

    for (int k = t; k < NCAND; k += 256) {
        float4 bb = s_bx[k];
        s_ar[k] = fmaxf(bb.z - bb.x, 0.0f) * fmaxf(bb.w - bb.y, 0.0f);
    }
    __syncthreads();

    float* orow = out + (size_t)p * MAXOUT * 5;

    for (int it = 0; it < MAXOUT; ++it) {
        // ---- argmax over live scores ----
        float ms = -1e30f; int mj = NCAND;
        for (int k = t; k < NCAND; k += 256) {
            float v = s_sc[k];
            if (v > ms) { ms = v; mj = k; }
        }
        // wave32 shuffle reduce (max, tie -> smaller index)
        for (int off = 16; off > 0; off >>= 1) {
            float os = __shfl_xor(ms, off, 32);
            int   oj = __shfl_xor(mj, off, 32);
            if (os > ms || (os == ms && oj < mj)) { ms = os; mj = oj; }
        }
        if ((t & 31) == 0) { rS[t >> 5] = ms; rJ[t >> 5] = mj; }
        __syncthreads();
        if (t == 0) {
            float bs = rS[0]; int bj = rJ[0];
            for (int w = 1; w < 8; ++w)
                if (rS[w] > bs || (rS[w] == bs && rJ[w] < bj)) { bs = rS[w]; bj = rJ[w]; }
            sBest = bs; sBestJ = bj;
        }
        __syncthreads();
        float bs = sBest; int bj = sBestJ;

        if (!(bs > 0.0f)) {
            // everything remaining is zero: fill and exit (uniform branch)
            float* z = orow + (size_t)it * 5;
            int rem = (MAXOUT - it) * 5;
            for (int q = t; q < rem; q += 256) z[q] = 0.0f;
            break;
        }

        float4 bb = s_bx[bj];
        if (t == 0) {
            float* o = orow + (size_t)it * 5;
            o[0] = bs; o[1] = bb.x; o[2] = bb.y; o[3] = bb.z; o[4] = bb.w;
        }
        float aj = s_ar[bj];
        for (int k = t; k < NCAND; k += 256) {
            float4 ob = s_bx[k];
            float xx1 = fmaxf(bb.x, ob.x);
            float yy1 = fmaxf(bb.y, ob.y);
            float xx2 = fminf(bb.z, ob.z);
            float yy2 = fminf(bb.w, ob.w);
            float inter = fmaxf(xx2 - xx1, 0.0f) * fmaxf(yy2 - yy1, 0.0f);
            float iou = inter / (aj + s_ar[k] - inter + 1e-8f);
            if (iou >= NMS_T) s_sc[k] = -1.0f;
        }
        __syncthreads();
    }
}

// ---------------------------------------------------------------------
extern "C" void kernel_launch(void* const* d_in, const int* in_sizes, int n_in,
                              void* d_out, int out_size, void* d_ws, size_t ws_size,
                              hipStream_t stream)
{
    const float* arm_cls = (const float*)d_in[0];
    const float* arm_loc = (const float*)d_in[1];
    const float* odm_cls = (const float*)d_in[2];
    const float* odm_loc = (const float*)d_in[3];
    const float* anchors = (const float*)d_in[4];
    float* out = (float*)d_out;

    // workspace layout (floats):
    //   boxes   : NB*NA*4          = 4,194,304
    //   scoresT : NB*NCLS*NA       = 20,971,520
    //   candS   : NPAIR*CANDPAD    =   163,840
    //   candB   : NPAIR*CANDPAD*4  =   655,360   (~99 MB total)
    float* boxes   = (float*)d_ws;
    float* scoresT = boxes   + (size_t)NB * NA * 4;
    float* candS   = scoresT + (size_t)NB * NCLS * NA;
    float* candB   = candS   + (size_t)NPAIR * CANDPAD;

    decode_kernel<<<(NB * NA) / 256, 256, 0, stream>>>(
        arm_cls, arm_loc, odm_cls, odm_loc, anchors, boxes, scoresT);
    select_kernel<<<NPAIR, 256, 0, stream>>>(scoresT, boxes, candS, candB);
    nms_kernel<<<NPAIR, 256, 0, stream>>>(candS, candB, out);
}